// SinkhornKnopp_4363686772738
// MI455X (gfx1250) — compile-verified
//
#include <hip/hip_runtime.h>
#include <math.h>

// Sinkhorn-Knopp via scaling vectors: y = diag(u) * E * diag(v), E = exp(x).
// Per iteration one fused kernel: rebuild v from previous partial colsums,
// row-GEMV -> update u, then WMMA-based partial col-GEMV for next v.
// Final pass materializes y in place in d_out (d_out holds E during iterations).

#define SK_B    32
#define SK_N    2048
#define SK_ITER 20
#define SK_EPS  1e-12f
#define RB      16      // row blocks per batch
#define RPB     128     // rows per block (RB*RPB == SK_N)
#define THREADS 256     // 8 wave32

typedef __attribute__((ext_vector_type(2))) float v2f;
typedef __attribute__((ext_vector_type(8))) float v8f;

__global__ __launch_bounds__(THREADS) void sk_exp(const float* __restrict__ x,
                                                  float* __restrict__ e) {
    size_t i = (size_t)blockIdx.x * THREADS + threadIdx.x;   // float4 index
    float4 a = ((const float4*)x)[i];
    float4 r;
    r.x = expf(a.x); r.y = expf(a.y); r.z = expf(a.z); r.w = expf(a.w);
    ((float4*)e)[i] = r;
}

__global__ __launch_bounds__(THREADS) void sk_init_u(float* __restrict__ u) {
    u[(size_t)blockIdx.x * THREADS + threadIdx.x] = 1.0f;
}

__global__ __launch_bounds__(THREADS) void sk_iter(
    const float* __restrict__ e, float* __restrict__ u,
    const float* __restrict__ v_prev, float* __restrict__ v_cur,
    const float* __restrict__ pc_prev, float* __restrict__ pc_cur, int first)
{
    __shared__ __align__(16) float v_lds[SK_N];
    __shared__ float uprime[RPB];

    const int tid  = threadIdx.x;
    const int blk  = blockIdx.x;          // 0..RB-1
    const int bat  = blockIdx.y;          // 0..SK_B-1
    const int lane = tid & 31;
    const int wave = tid >> 5;

    // ---- phase 0: reconstruct current v (redundant per WG; blk 0 persists it)
    if (first) {
        for (int j = tid; j < SK_N; j += THREADS) {
            v_lds[j] = 1.0f;
            if (blk == 0) v_cur[(size_t)bat * SK_N + j] = 1.0f;
        }
    } else {
        const float* pcb = pc_prev + (size_t)bat * RB * SK_N;
        for (int j = tid; j < SK_N; j += THREADS) {
            float s = 0.f;
            #pragma unroll
            for (int b = 0; b < RB; ++b) s += pcb[(size_t)b * SK_N + j];
            float vp = v_prev[(size_t)bat * SK_N + j];
            float vn = vp / fmaxf(vp * s, SK_EPS);
            v_lds[j] = vn;
            if (blk == 0) v_cur[(size_t)bat * SK_N + j] = vn;
        }
    }
    __syncthreads();

    // ---- phase 1: row sums s_i = sum_j E_ij * v_j ; update u in place
    const float* eblk = e + ((size_t)bat * SK_N + (size_t)blk * RPB) * SK_N;
    for (int r = wave; r < RPB; r += 8) {
        const float* row = eblk + (size_t)r * SK_N;
        float4 acc = {0.f, 0.f, 0.f, 0.f};
        #pragma unroll 4
        for (int c = 0; c < SK_N / 128; ++c) {
            int j = (lane << 2) + (c << 7);
            if (c == 0 && r + 8 < RPB)
                __builtin_prefetch(row + 8 * SK_N + j, 0, 0);  // global_prefetch_b8
            float4 ev = *(const float4*)(row + j);
            float4 vv = *(const float4*)(v_lds + j);
            acc.x = fmaf(ev.x, vv.x, acc.x);
            acc.y = fmaf(ev.y, vv.y, acc.y);
            acc.z = fmaf(ev.z, vv.z, acc.z);
            acc.w = fmaf(ev.w, vv.w, acc.w);
        }
        float s = (acc.x + acc.y) + (acc.z + acc.w);
        #pragma unroll
        for (int m = 16; m > 0; m >>= 1) s += __shfl_xor(s, m, 32);
        if (lane == 0) {
            size_t gi = (size_t)bat * SK_N + (size_t)blk * RPB + r;
            float uo = u[gi];
            float un = uo / fmaxf(uo * s, SK_EPS);
            uprime[r] = un;
            u[gi] = un;
        }
    }
    __syncthreads();

    // ---- phase 2: partial column sums pc_j = sum_{r in block} u'_r * E_rj
    // via V_WMMA_F32_16X16X4_F32: A(16x4) = u' chunk broadcast over M rows,
    // B(4x16) = E tile (lanes 0-15: K=0/1, lanes 16-31: K=2/3 per ISA layout),
    // D rows are identical == partial column sums for 16 columns.
    const int half = lane >> 4;
    const int ln   = lane & 15;
    float* pcout = pc_cur + ((size_t)bat * RB + blk) * SK_N + wave * 256;
    for (int t = 0; t < 16; ++t) {
        const int j0 = wave * 256 + t * 16;
        v8f C = {0.f, 0.f, 0.f, 0.f, 0.f, 0.f, 0.f, 0.f};
        for (int k0 = 0; k0 < RPB; k0 += 4) {
            const int rA = k0 + (half << 1);
            v2f A, Bm;
            A.x  = uprime[rA];
            A.y  = uprime[rA + 1];
            Bm.x = eblk[(size_t)rA * SK_N + j0 + ln];
            Bm.y = eblk[(size_t)(rA + 1) * SK_N + j0 + ln];
            C = __builtin_amdgcn_wmma_f32_16x16x4_f32(
                    false, A, false, Bm, (short)0, C, false, false);
        }
        if (half == 0) pcout[t * 16 + ln] = C[0];   // D[M=0][N=ln]
    }
}

__global__ __launch_bounds__(THREADS) void sk_vfin(
    const float* __restrict__ v_last, const float* __restrict__ pc_last,
    float* __restrict__ v_fin)
{
    size_t idx = (size_t)blockIdx.x * THREADS + threadIdx.x;  // over B*N
    size_t bat = idx / SK_N;
    size_t j   = idx % SK_N;
    const float* pcb = pc_last + bat * RB * SK_N;
    float s = 0.f;
    #pragma unroll
    for (int b = 0; b < RB; ++b) s += pcb[(size_t)b * SK_N + j];
    float vp = v_last[idx];
    v_fin[idx] = vp / fmaxf(vp * s, SK_EPS);
}

__global__ __launch_bounds__(THREADS) void sk_final(
    float* __restrict__ y, const float* __restrict__ u, const float* __restrict__ v)
{
    size_t f   = (size_t)blockIdx.x * THREADS + threadIdx.x;  // float4 index
    size_t c4  = f & 511;                 // N/4 = 512
    size_t row = (f >> 9) & (SK_N - 1);
    size_t bat = f >> 20;                 // 512*2048 = 2^20 float4 per batch
    float4 y4 = ((float4*)y)[f];
    float ui = u[bat * SK_N + row];
    float4 v4 = *(const float4*)(v + bat * SK_N + c4 * 4);
    y4.x *= ui * v4.x; y4.y *= ui * v4.y;
    y4.z *= ui * v4.z; y4.w *= ui * v4.w;
    ((float4*)y)[f] = y4;
}

extern "C" void kernel_launch(void* const* d_in, const int* in_sizes, int n_in,
                              void* d_out, int out_size, void* d_ws, size_t ws_size,
                              hipStream_t stream) {
    (void)in_sizes; (void)n_in; (void)out_size; (void)ws_size;
    const float* x = (const float*)d_in[0];
    float* y = (float*)d_out;                 // holds E during iterations
    float* ws = (float*)d_ws;                 // ~8.75 MB used
    float* u   = ws;                          // B*N
    float* v0  = u  + (size_t)SK_B * SK_N;    // B*N
    float* v1  = v0 + (size_t)SK_B * SK_N;    // B*N
    float* pc0 = v1 + (size_t)SK_B * SK_N;    // B*RB*N
    float* pc1 = pc0 + (size_t)SK_B * RB * SK_N;

    const size_t total4 = (size_t)SK_B * SK_N * SK_N / 4;
    sk_exp  <<<dim3((unsigned)(total4 / THREADS)), THREADS, 0, stream>>>(x, y);
    sk_init_u<<<dim3(SK_B * SK_N / THREADS), THREADS, 0, stream>>>(u);

    for (int i = 0; i < SK_ITER; ++i) {
        float* vc  = (i & 1) ? v1  : v0;
        float* vp  = (i & 1) ? v0  : v1;
        float* pcc = (i & 1) ? pc1 : pc0;
        float* pcp = (i & 1) ? pc0 : pc1;
        sk_iter<<<dim3(RB, SK_B), THREADS, 0, stream>>>(
            y, u, vp, vc, pcp, pcc, (i == 0) ? 1 : 0);
    }
    // After i=19 (odd): v_cur = v1 (holds v_19), pc_cur = pc1 (partials for v_20).
    sk_vfin <<<dim3(SK_B * SK_N / THREADS), THREADS, 0, stream>>>(v1, pc1, v0);
    sk_final<<<dim3((unsigned)(total4 / THREADS)), THREADS, 0, stream>>>(y, u, v0);
}